// GHMC_Loss_46686294508029
// MI455X (gfx1250) — compile-verified
//
#include <hip/hip_runtime.h>
#include <cstdint>

#define BINS 10
#define TPB 256
#define NWAVE (TPB / 32)
#define NBLOCKS 2048
#define HPAD 16     // padded bin stride so counts/sums land in distinct LDS banks
#define PSTRIDE 32  // padded per-block partial row (20 used of 32 floats)

typedef __attribute__((ext_vector_type(4))) float v4f;
typedef __attribute__((ext_vector_type(4))) int v4i;

// CDNA5 async global->LDS copy path (builtin confirmed present on this
// toolchain; source param is an int4 AS1 pointer, dest is AS3).
#if defined(__AMDGCN__) &&                                                     \
    __has_builtin(__builtin_amdgcn_global_load_async_to_lds_b128) &&           \
    __has_builtin(__builtin_amdgcn_s_wait_asynccnt)
#define USE_ASYNC 1
#else
#define USE_ASYNC 0
#endif

// Per-element update: bin index + BCE, accumulated into this wave's private
// LDS histogram. whist[0..9] = counts, whist[HPAD..HPAD+9] = bce sums.
__device__ __forceinline__ void accum_elem(float* whist, float p, float t) {
  float g = fabsf(p - t);                  // in [0,1)
  int b = (int)(g * 10.0f);                // == floor, g >= 0
  b = b > (BINS - 1) ? (BINS - 1) : b;
  // t is exactly 0.0 or 1.0 -> bce = -log(p) or -log(1-p)
  float x = (t > 0.5f) ? p : (1.0f - p);
  float bce = -__logf(x);                  // v_log_f32 (TRANS, co-executes)
  atomicAdd(&whist[b], 1.0f);              // ds_add_f32, bank = b
  atomicAdd(&whist[HPAD + b], bce);        // ds_add_f32, bank = 16+b
}

__global__ void ghm_init(float* __restrict__ gpart) {
  int i = threadIdx.x;
  if (i < PSTRIDE) gpart[i] = 0.0f;
}

__global__ __launch_bounds__(TPB) void ghm_main(const float* __restrict__ pin,
                                                const float* __restrict__ tin,
                                                float* __restrict__ gpart,
                                                long long n, int det) {
  __shared__ float hist[NWAVE * 2 * HPAD];  // 8 waves * (counts|sums) * 16
#if USE_ASYNC
  __shared__ v4f sp[2][TPB];
  __shared__ v4f st[2][TPB];
#endif
  const int tid = threadIdx.x;
  hist[tid] = 0.0f;  // NWAVE*2*HPAD == TPB == 256
  __syncthreads();
  float* whist = &hist[(tid >> 5) * 2 * HPAD];

  const long long n4 = n >> 2;
  const long long gtid = (long long)blockIdx.x * TPB + tid;
  const long long stride = (long long)gridDim.x * TPB;
  const v4f* p4 = (const v4f*)pin;
  const v4f* t4 = (const v4f*)tin;

#if USE_ASYNC
  // Double-buffered async global->LDS staging (each lane stages its own 16B
  // per array; ASYNCcnt +2 per stage). In-order completion => wait<=2 means
  // the older stage has landed in LDS.
  auto issue = [&](long long idx, int buf) {
    __builtin_amdgcn_global_load_async_to_lds_b128(
        (__attribute__((address_space(1))) v4i*)(uintptr_t)(p4 + idx),
        (__attribute__((address_space(3))) v4i*)(uint32_t)(uintptr_t) &
            sp[buf][tid],
        0, 0);
    __builtin_amdgcn_global_load_async_to_lds_b128(
        (__attribute__((address_space(1))) v4i*)(uintptr_t)(t4 + idx),
        (__attribute__((address_space(3))) v4i*)(uint32_t)(uintptr_t) &
            st[buf][tid],
        0, 0);
  };
  if (gtid < n4) issue(gtid, 0);
  if (gtid + stride < n4) issue(gtid + stride, 1);
  int buf = 0;
  for (long long cur = gtid; cur < n4; cur += stride) {
    if (cur + stride < n4)
      __builtin_amdgcn_s_wait_asynccnt(2);  // stage 'cur' complete
    else
      __builtin_amdgcn_s_wait_asynccnt(0);  // tail: drain everything
    asm volatile("" ::: "memory");          // keep LDS reads below the wait
    v4f pv = sp[buf][tid];
    v4f tv = st[buf][tid];
    accum_elem(whist, pv.x, tv.x);
    accum_elem(whist, pv.y, tv.y);
    accum_elem(whist, pv.z, tv.z);
    accum_elem(whist, pv.w, tv.w);
    long long nxt = cur + 2 * stride;
    if (nxt < n4) {
      // our ds_load of this buffer must retire before the async write reuses it
      asm volatile("s_wait_dscnt 0x0" ::: "memory");
      issue(nxt, buf);
    }
    buf ^= 1;
  }
#else
  for (long long cur = gtid; cur < n4; cur += stride) {
    // read-once stream (>192MB L2): non-temporal b128 loads
    v4f pv = __builtin_nontemporal_load(p4 + cur);
    v4f tv = __builtin_nontemporal_load(t4 + cur);
    accum_elem(whist, pv.x, tv.x);
    accum_elem(whist, pv.y, tv.y);
    accum_elem(whist, pv.z, tv.z);
    accum_elem(whist, pv.w, tv.w);
  }
#endif

  // scalar tail for n % 4 (none for this shape, kept for generality)
  for (long long j = (n4 << 2) + gtid; j < n; j += stride) {
    accum_elem(whist, pin[j], tin[j]);
  }

  __syncthreads();
  // fold the 8 wave histograms into 20 block partials
  if (tid < 2 * BINS) {
    int which = tid / BINS;  // 0 = counts, 1 = bce sums
    int b = tid - which * BINS;
    float acc = 0.0f;
#pragma unroll
    for (int w = 0; w < NWAVE; ++w) acc += hist[w * 2 * HPAD + which * HPAD + b];
    if (det) {
      // deterministic: private row per block, reduced in fixed order later
      gpart[(long long)blockIdx.x * PSTRIDE + tid] = acc;
    } else {
      atomicAdd(&gpart[tid], acc);
    }
  }
}

// Reduce nrows x 20 partials (row stride PSTRIDE) in a fixed order, then
// loss = (1/nbins) * sum_b bce_sum[b] / counts[b] over non-empty bins.
__global__ __launch_bounds__(256) void ghm_finalize(
    const float* __restrict__ gpart, float* __restrict__ out, int nrows) {
  __shared__ float red[8][32];
  const int t = threadIdx.x;
  const int j = t & 31;   // column (20 valid)
  const int r0 = t >> 5;  // row group 0..7
  float acc = 0.0f;
  if (j < 2 * BINS) {
    for (int k = r0; k < nrows; k += 8) acc += gpart[(long long)k * PSTRIDE + j];
  }
  red[r0][j] = acc;
  __syncthreads();
  if (t == 0) {
    float cs[2 * BINS];
    for (int jj = 0; jj < 2 * BINS; ++jj) {
      float s = 0.0f;
#pragma unroll
      for (int r = 0; r < 8; ++r) s += red[r][jj];
      cs[jj] = s;
    }
    float loss = 0.0f;
    int nb = 0;
    for (int b = 0; b < BINS; ++b) {
      if (cs[b] > 0.0f) {
        nb += 1;
        loss += cs[BINS + b] / cs[b];
      }
    }
    out[0] = loss / (nb > 0 ? (float)nb : 1.0f);
  }
}

extern "C" void kernel_launch(void* const* d_in, const int* in_sizes, int n_in,
                              void* d_out, int out_size, void* d_ws,
                              size_t ws_size, hipStream_t stream) {
  (void)n_in;
  (void)out_size;
  const float* pin = (const float*)d_in[0];
  const float* tin = (const float*)d_in[1];
  float* gpart = (float*)d_ws;
  float* out = (float*)d_out;
  long long n = (long long)in_sizes[0];

  const size_t det_bytes = (size_t)NBLOCKS * PSTRIDE * sizeof(float);  // 256 KB
  const int det = (ws_size >= det_bytes) ? 1 : 0;

  if (!det) ghm_init<<<1, 32, 0, stream>>>(gpart);
  ghm_main<<<NBLOCKS, TPB, 0, stream>>>(pin, tin, gpart, n, det);
  ghm_finalize<<<1, 256, 0, stream>>>(gpart, out, det ? NBLOCKS : 1);
}